// SpatialEncoding_4157528343275
// MI455X (gfx1250) — compile-verified
//
#include <hip/hip_runtime.h>

// Spatial encoding scatter for gfx1250 (MI455X).
//
// reference: out = zeros(8192,8192); out[src,dst] = b[clip(min(path_len,5)-1,0,4)]
//
// Pure store-bandwidth problem (~262 MB written, ~6 MB read, ~0 FLOPs).
// Floor at 23.3 TB/s HBM ~= 11.5 us. No matmul -> WMMA not applicable;
// the CDNA5 levers that matter are b128 stores + TH_STORE_NT streaming
// hints (output is 256 MB > 192 MB L2 and is never re-read on device).

typedef __attribute__((ext_vector_type(4))) float v4f;

#define N_NODES  8192
#define N_PAIRS  524288
#define MAX_PATH 5

// ---------------------------------------------------------------------------
// Kernel 1: streaming zero-fill of the 8192x8192 f32 output.
// float4 (b128) stores, nontemporal -> global_store_b128 ... th:TH_STORE_NT.
// Grid-stride so any launch size saturates all WGPs.
// ---------------------------------------------------------------------------
__global__ void spatial_zero_fill(float* __restrict__ out, long total4) {
  long i      = (long)blockIdx.x * blockDim.x + threadIdx.x;
  long stride = (long)gridDim.x * blockDim.x;
  v4f z = {0.0f, 0.0f, 0.0f, 0.0f};
  v4f* __restrict__ o = (v4f*)out;
  for (; i < total4; i += stride) {
    __builtin_nontemporal_store(z, o + i);
  }
}

// ---------------------------------------------------------------------------
// Kernel 2: scatter b[bucket] into out[src*8192 + dst].
// 3 coalesced NT loads + 1 scattered NT store per pair.
// The 5-entry bias table goes through the scalar path (uniform pointer,
// constant indices -> s_load) and is selected with v_cndmask, so no
// per-lane gather from memory is needed.
// ---------------------------------------------------------------------------
__global__ void spatial_scatter(const float* __restrict__ b,
                                const int*   __restrict__ src,
                                const int*   __restrict__ dst,
                                const int*   __restrict__ plen,
                                float*       __restrict__ out) {
  int i = blockIdx.x * blockDim.x + threadIdx.x;
  if (i >= N_PAIRS) return;

  // scalar loads of the tiny bias table (uniform across the wave)
  float b0 = b[0], b1 = b[1], b2 = b[2], b3 = b[3], b4 = b[4];

  int s  = __builtin_nontemporal_load(src  + i);
  int d  = __builtin_nontemporal_load(dst  + i);
  int pl = __builtin_nontemporal_load(plen + i);

  // bucket = clip(min(pl, MAX_PATH) - 1, 0, MAX_PATH-1)
  int t = pl < MAX_PATH ? pl : MAX_PATH;
  t = t - 1;
  t = t < 0 ? 0 : t;   // t in [0,4]

  float v = (t == 0) ? b0
          : (t == 1) ? b1
          : (t == 2) ? b2
          : (t == 3) ? b3
          :            b4;

  long off = (long)s * N_NODES + d;
  __builtin_nontemporal_store(v, out + off);
}

extern "C" void kernel_launch(void* const* d_in, const int* in_sizes, int n_in,
                              void* d_out, int out_size, void* d_ws, size_t ws_size,
                              hipStream_t stream) {
  // setup_inputs() order: x (unused, shape only), b, src, dst, path_len
  const float* b    = (const float*)d_in[1];
  const int*   src  = (const int*)  d_in[2];
  const int*   dst  = (const int*)  d_in[3];
  const int*   plen = (const int*)  d_in[4];
  float*       out  = (float*)d_out;

  // 8192*8192 floats = 16M float4 stores
  const long total4 = (long)N_NODES * N_NODES / 4;
  const int  zthreads = 256;
  const int  zblocks  = 8192;   // 2M threads, 8 b128 stores each (grid-stride)
  spatial_zero_fill<<<zblocks, zthreads, 0, stream>>>(out, total4);

  const int sthreads = 256;
  const int sblocks  = (N_PAIRS + sthreads - 1) / sthreads;
  spatial_scatter<<<sblocks, sthreads, 0, stream>>>(b, src, dst, plen, out);
}